// EnergyModel_46566035423670
// MI455X (gfx1250) — compile-verified
//
#include <hip/hip_runtime.h>
#include <math.h>

// MACE-style 2-layer GNN energy+forces for gfx1250 (MI455X).
//  - wave-per-edge message kernels (wave32), W_r staged in LDS, f32 global atomics
//  - dense F x F update GEMMs on V_WMMA_F32_16X16X4_F32 (exact fp32 math)
//  - hand-derived reverse pass for forces; layer-1 upstream grad collapses to a
//    constant vector v1 = W_out[1] @ w_read[1], so only ONE backward GEMM needed.

#define F 128
#define NB 8
#define RMAX 5.0f

typedef __attribute__((ext_vector_type(2))) float v2f;
typedef __attribute__((ext_vector_type(8))) float v8f;

__device__ __forceinline__ float waveReduceSum(float v) {
#pragma unroll
  for (int m = 16; m >= 1; m >>= 1) v += __shfl_xor(v, m, 32);
  return v;
}

// ---------------------------------------------------------------- utilities
__global__ void k_zero(float* __restrict__ p, size_t n) {
  size_t i = (size_t)blockIdx.x * blockDim.x + threadIdx.x;
  size_t st = (size_t)gridDim.x * blockDim.x;
  for (; i < n; i += st) p[i] = 0.0f;
}

__global__ void k_ginit(float* __restrict__ g, const float* __restrict__ wr0, size_t n) {
  size_t i = (size_t)blockIdx.x * blockDim.x + threadIdx.x;
  size_t st = (size_t)gridDim.x * blockDim.x;
  for (; i < n; i += st) g[i] = wr0[i & (F - 1)];
}

// v1[f] = sum_j W_out1[f,j] * w_read1[j]
__global__ void k_v1(const float* __restrict__ Wout1, const float* __restrict__ wr1,
                     float* __restrict__ v1) {
  int f = threadIdx.x;
  float acc = 0.0f;
  for (int j = 0; j < F; ++j) acc += Wout1[(size_t)f * F + j] * wr1[j];
  v1[f] = acc;
}

// ------------------------------------------- node linear: out = attrs @ W (10xF)
// optional e0 readout: energy[batch[n]] += attrs . ae
__global__ void k_node_lin(const float* __restrict__ attrs, const float* __restrict__ W,
                           float* __restrict__ out, const float* __restrict__ ae,
                           const int* __restrict__ batch, float* __restrict__ energy,
                           int Nn) {
  __shared__ float sW[10 * F];
  for (int i = threadIdx.x; i < 10 * F; i += blockDim.x) sW[i] = W[i];
  __syncthreads();
  int lane = threadIdx.x & 31;
  int n = blockIdx.x * 8 + (threadIdx.x >> 5);
  if (n >= Nn) return;
  float av = (lane < 10) ? attrs[(size_t)n * 10 + lane] : 0.0f;
  int f0 = lane * 4;
  float o0 = 0, o1 = 0, o2 = 0, o3 = 0;
#pragma unroll
  for (int k = 0; k < 10; ++k) {
    float a = __shfl(av, k, 32);
    const float* w = &sW[k * F + f0];
    o0 += a * w[0]; o1 += a * w[1]; o2 += a * w[2]; o3 += a * w[3];
  }
  float* op = out + (size_t)n * F + f0;
  op[0] = o0; op[1] = o1; op[2] = o2; op[3] = o3;
  if (ae) {
    float e = (lane < 10) ? av * ae[lane] : 0.0f;
    e = waveReduceSum(e);
    if (lane == 0) atomicAdd(&energy[batch[n]], e);
  }
}

// ------------------------------------------------------- edge geometry (fwd)
__global__ void k_edge_geom(const float* __restrict__ pos, const float* __restrict__ shifts,
                            const int* __restrict__ ei, const float* __restrict__ w_sh,
                            float* __restrict__ evec, float* __restrict__ er,
                            float* __restrict__ eeb, float* __restrict__ es, int E) {
  int e = blockIdx.x * blockDim.x + threadIdx.x;
  if (e >= E) return;
  int snd = ei[e], rcv = ei[E + e];
  float vx = pos[rcv * 3 + 0] - pos[snd * 3 + 0] + shifts[e * 3 + 0];
  float vy = pos[rcv * 3 + 1] - pos[snd * 3 + 1] + shifts[e * 3 + 1];
  float vz = pos[rcv * 3 + 2] - pos[snd * 3 + 2] + shifts[e * 3 + 2];
  float r = sqrtf(vx * vx + vy * vy + vz * vz + 1e-12f);
  float inv = 1.0f / r;
  float x = vx * inv, y = vy * inv, z = vz * inv;
  float x2 = x * x, y2 = y * y, z2 = z * z;
  const float S3 = 1.7320508f, S5 = 2.2360680f, S15 = 3.8729833f;
  const float C1 = 2.0916500f, C2 = 10.2469508f, C3 = 1.6201852f, C4 = 1.3228757f,
              C5 = 5.1234754f;
  float Y[16];
  Y[0] = 1.0f;            Y[1] = S3 * x;          Y[2] = S3 * y;          Y[3] = S3 * z;
  Y[4] = S15 * x * y;     Y[5] = S15 * y * z;     Y[6] = 0.5f * S5 * (3.f * z2 - 1.f);
  Y[7] = S15 * x * z;     Y[8] = 0.5f * S15 * (x2 - y2);
  Y[9] = C1 * y * (3.f * x2 - y2);  Y[10] = C2 * x * y * z;
  Y[11] = C3 * y * (5.f * z2 - 1.f); Y[12] = C4 * z * (5.f * z2 - 3.f);
  Y[13] = C3 * x * (5.f * z2 - 1.f); Y[14] = C5 * z * (x2 - y2);
  Y[15] = C1 * x * (x2 - 3.f * y2);
  float s0 = 0, s1 = 0;
#pragma unroll
  for (int c = 0; c < 16; ++c) { s0 += Y[c] * w_sh[c]; s1 += Y[c] * w_sh[16 + c]; }
  // radial bessel * envelope (p=6): env = 1 - 28u^6 + 48u^7 - 21u^8
  float mask = (r < RMAX) ? 1.0f : 0.0f;
  float ur = r * (1.0f / RMAX);
  float u2 = ur * ur, u6 = u2 * u2 * u2, u7 = u6 * ur, u8 = u7 * ur;
  float env = (1.0f - 28.0f * u6 + 48.0f * u7 - 21.0f * u8) * mask;
  const float K0 = 0.632455532f;      // sqrt(2/RMAX)
  const float PIoR = 0.6283185307f;   // pi/RMAX
#pragma unroll
  for (int k = 1; k <= NB; ++k)
    eeb[(size_t)e * NB + k - 1] = K0 * sinf((float)k * PIoR * r) * inv * env;
  evec[e * 3 + 0] = vx; evec[e * 3 + 1] = vy; evec[e * 3 + 2] = vz;
  er[e] = r;
  es[(size_t)e * 2 + 0] = s0; es[(size_t)e * 2 + 1] = s1;
}

// ------------------------------------------- forward message + scatter (wave/edge)
__global__ void k_msg(const float* __restrict__ h, const float* __restrict__ eeb,
                      const float* __restrict__ es, const int* __restrict__ ei,
                      const float* __restrict__ Wr, float* __restrict__ agg, int E,
                      int layer) {
  __shared__ float sWr[NB * F];
  for (int i = threadIdx.x; i < NB * F; i += blockDim.x) sWr[i] = Wr[i];
  __syncthreads();
  int lane = threadIdx.x & 31;
  int e = blockIdx.x * 8 + (threadIdx.x >> 5);
  if (e >= E) return;
  int snd = ei[e], rcv = ei[E + e];
  float se = es[(size_t)e * 2 + layer];
  float ebl = (lane < NB) ? eeb[(size_t)e * NB + lane] : 0.0f;
  int f0 = lane * 4;
  __builtin_prefetch(h + (size_t)snd * F + f0, 0, 0);
  const float4 hs = *(const float4*)(h + (size_t)snd * F + f0);
  float R0 = 0, R1 = 0, R2 = 0, R3 = 0;
#pragma unroll
  for (int k = 0; k < NB; ++k) {
    float eb = __shfl(ebl, k, 32);
    const float* w = &sWr[k * F + f0];
    R0 += eb * w[0]; R1 += eb * w[1]; R2 += eb * w[2]; R3 += eb * w[3];
  }
  float* ap = agg + (size_t)rcv * F + f0;
  atomicAdd(ap + 0, hs.x * R0 * se);
  atomicAdd(ap + 1, hs.y * R1 * se);
  atomicAdd(ap + 2, hs.z * R2 * se);
  atomicAdd(ap + 3, hs.w * R3 * se);
}

// ------------------------------------------- fp32 WMMA GEMM: D = A(NxF) @ B(FxF) [+C]
// one wave computes a 16x16 C tile; block = 8 waves = 8 column tiles (F=128).
template <bool TRANS_B, bool LOAD_C>
__global__ void k_gemm_wmma(const float* __restrict__ A, const float* __restrict__ B,
                            const float* __restrict__ Cin, float* __restrict__ D,
                            int Nrows) {
  int wv = threadIdx.x >> 5, lane = threadIdx.x & 31;
  int m0 = blockIdx.x * 16;
  int n0 = wv * 16;
  int col = lane & 15;
  int hi = lane >> 4;
  int rowA = m0 + col;  // A fragment rows live in lane&15
  int rowAc = rowA < Nrows ? rowA : Nrows - 1;
  v8f c;
#pragma unroll
  for (int r = 0; r < 8; ++r) {
    if (LOAD_C) {
      int m = m0 + r + 8 * hi;
      int mc = m < Nrows ? m : Nrows - 1;
      c[r] = Cin[(size_t)mc * F + n0 + col];
    } else {
      c[r] = 0.0f;
    }
  }
#pragma unroll 8
  for (int k0 = 0; k0 < F; k0 += 4) {
    // A 16x4 f32 layout: lanes 0-15 -> K=k0,k0+1 ; lanes 16-31 -> K=k0+2,k0+3
    const float* ap = A + (size_t)rowAc * F + k0 + hi * 2;
    v2f a; a.x = ap[0]; a.y = ap[1];
    v2f b;
    if (!TRANS_B) {
      b.x = B[(size_t)(k0 + hi * 2 + 0) * F + n0 + col];
      b.y = B[(size_t)(k0 + hi * 2 + 1) * F + n0 + col];
    } else {  // B = W^T  =>  B[k][n] = W[n][k]
      const float* bp = B + (size_t)(n0 + col) * F + k0 + hi * 2;
      b.x = bp[0]; b.y = bp[1];
    }
    c = __builtin_amdgcn_wmma_f32_16x16x4_f32(false, a, false, b, (short)0, c, false,
                                              false);
  }
#pragma unroll
  for (int r = 0; r < 8; ++r) {
    int m = m0 + r + 8 * hi;
    if (m < Nrows) D[(size_t)m * F + n0 + col] = c[r];
  }
}

// ------------------------------------------- readout: energy[batch[n]] += h[n].w_read
__global__ void k_readout(const float* __restrict__ h, const float* __restrict__ wr,
                          const int* __restrict__ batch, float* __restrict__ energy,
                          int Nn) {
  int lane = threadIdx.x & 31;
  int n = blockIdx.x * 8 + (threadIdx.x >> 5);
  if (n >= Nn) return;
  const float4 hv = *(const float4*)(h + (size_t)n * F + lane * 4);
  const float4 wv = *(const float4*)(wr + lane * 4);
  float d = hv.x * wv.x + hv.y * wv.y + hv.z * wv.z + hv.w * wv.w;
  d = waveReduceSum(d);
  if (lane == 0) atomicAdd(&energy[batch[n]], d);
}

// ------------------------------------------- edge backward (per layer, wave/edge)
// LAYER==1: upstream = v1 (const), scatter g_h1[snd] += v1*R1*s1, geb '=' write
// LAYER==0: upstream = g_agg0[rcv], geb '+=' accumulate
template <int LAYER>
__global__ void k_edge_bwd(const float* __restrict__ h, const float* __restrict__ eeb,
                           const float* __restrict__ es, const int* __restrict__ ei,
                           const float* __restrict__ Wr, const float* __restrict__ v1,
                           const float* __restrict__ gagg, float* __restrict__ gh,
                           float* __restrict__ geb, float* __restrict__ gs, int E) {
  __shared__ float sWr[NB * F];
  __shared__ float sv1[F];
  for (int i = threadIdx.x; i < NB * F; i += blockDim.x) sWr[i] = Wr[i];
  if (LAYER == 1) {
    if (threadIdx.x < F) sv1[threadIdx.x] = v1[threadIdx.x];
  }
  __syncthreads();
  int lane = threadIdx.x & 31;
  int e = blockIdx.x * 8 + (threadIdx.x >> 5);
  if (e >= E) return;
  int snd = ei[e];
  float se = es[(size_t)e * 2 + LAYER];
  float ebl = (lane < NB) ? eeb[(size_t)e * NB + lane] : 0.0f;
  int f0 = lane * 4;
  float R0 = 0, R1 = 0, R2 = 0, R3 = 0;
#pragma unroll
  for (int k = 0; k < NB; ++k) {
    float eb = __shfl(ebl, k, 32);
    const float* w = &sWr[k * F + f0];
    R0 += eb * w[0]; R1 += eb * w[1]; R2 += eb * w[2]; R3 += eb * w[3];
  }
  const float4 hs = *(const float4*)(h + (size_t)snd * F + f0);
  float g0, g1, g2, g3;
  if (LAYER == 1) {
    g0 = sv1[f0 + 0]; g1 = sv1[f0 + 1]; g2 = sv1[f0 + 2]; g3 = sv1[f0 + 3];
    float* gp = gh + (size_t)snd * F + f0;
    atomicAdd(gp + 0, g0 * R0 * se);
    atomicAdd(gp + 1, g1 * R1 * se);
    atomicAdd(gp + 2, g2 * R2 * se);
    atomicAdd(gp + 3, g3 * R3 * se);
  } else {
    int rcv = ei[E + e];
    const float4 ga = *(const float4*)(gagg + (size_t)rcv * F + f0);
    g0 = ga.x; g1 = ga.y; g2 = ga.z; g3 = ga.w;
  }
  // g_R = g * h[snd] * s ; g_eb[k] = sum_f g_R[f] * Wr[k,f] ; g_s = sum_f g*h*R
  float gR0 = g0 * hs.x * se, gR1 = g1 * hs.y * se, gR2 = g2 * hs.z * se,
        gR3 = g3 * hs.w * se;
  float gk[NB];
#pragma unroll
  for (int k = 0; k < NB; ++k) {
    const float* w = &sWr[k * F + f0];
    gk[k] = gR0 * w[0] + gR1 * w[1] + gR2 * w[2] + gR3 * w[3];
  }
  float gsp = g0 * hs.x * R0 + g1 * hs.y * R1 + g2 * hs.z * R2 + g3 * hs.w * R3;
#pragma unroll
  for (int m = 16; m >= 1; m >>= 1) {
#pragma unroll
    for (int k = 0; k < NB; ++k) gk[k] += __shfl_xor(gk[k], m, 32);
    gsp += __shfl_xor(gsp, m, 32);
  }
  if (lane == 0) {
    float* gp = geb + (size_t)e * NB;
#pragma unroll
    for (int k = 0; k < NB; ++k) {
      if (LAYER == 1) gp[k] = gk[k];
      else gp[k] += gk[k];
    }
    gs[(size_t)e * 2 + LAYER] = gsp;
  }
}

// ------------------------------------------- geometry backward -> forces
__global__ void k_geom_bwd(const float* __restrict__ evec, const float* __restrict__ er,
                           const float* __restrict__ geb, const float* __restrict__ gs,
                           const float* __restrict__ w_sh, const int* __restrict__ ei,
                           float* __restrict__ forces, int E) {
  int e = blockIdx.x * blockDim.x + threadIdx.x;
  if (e >= E) return;
  float vx = evec[e * 3 + 0], vy = evec[e * 3 + 1], vz = evec[e * 3 + 2];
  float r = er[e];
  float inv = 1.0f / r;
  float x = vx * inv, y = vy * inv, z = vz * inv;
  float x2 = x * x, y2 = y * y, z2 = z * z;
  float gs0 = gs[(size_t)e * 2 + 0], gs1 = gs[(size_t)e * 2 + 1];
  float gY[16];
#pragma unroll
  for (int c = 0; c < 16; ++c) gY[c] = gs0 * w_sh[c] + gs1 * w_sh[16 + c];
  const float S3 = 1.7320508f, S5 = 2.2360680f, S15 = 3.8729833f;
  const float C1 = 2.0916500f, C2 = 10.2469508f, C3 = 1.6201852f, C4 = 1.3228757f,
              C5 = 5.1234754f;
  float gux = S3 * gY[1] + S15 * y * gY[4] + S15 * z * gY[7] + S15 * x * gY[8] +
              6.f * C1 * x * y * gY[9] + C2 * y * z * gY[10] +
              C3 * (5.f * z2 - 1.f) * gY[13] + 2.f * C5 * x * z * gY[14] +
              3.f * C1 * (x2 - y2) * gY[15];
  float guy = S3 * gY[2] + S15 * x * gY[4] + S15 * z * gY[5] - S15 * y * gY[8] +
              3.f * C1 * (x2 - y2) * gY[9] + C2 * x * z * gY[10] +
              C3 * (5.f * z2 - 1.f) * gY[11] - 2.f * C5 * y * z * gY[14] -
              6.f * C1 * x * y * gY[15];
  float guz = S3 * gY[3] + S15 * y * gY[5] + 3.f * S5 * z * gY[6] + S15 * x * gY[7] +
              C2 * x * y * gY[10] + 10.f * C3 * y * z * gY[11] +
              C4 * (15.f * z2 - 3.f) * gY[12] + 10.f * C3 * x * z * gY[13] +
              C5 * (x2 - y2) * gY[14];
  // radial chain: d(eb_k)/dr = bess'_k * env + bess_k * env'
  float mask = (r < RMAX) ? 1.0f : 0.0f;
  float ur = r * (1.0f / RMAX);
  float u2 = ur * ur, u5 = u2 * u2 * ur, u6 = u5 * ur, u7 = u6 * ur, u8 = u7 * ur;
  float env = (1.0f - 28.0f * u6 + 48.0f * u7 - 21.0f * u8) * mask;
  float denv = (-168.0f * u5 + 336.0f * u6 - 168.0f * u7) * (1.0f / RMAX) * mask;
  const float K0 = 0.632455532f;
  const float PIoR = 0.6283185307f;
  float gr = 0.0f;
#pragma unroll
  for (int k = 1; k <= NB; ++k) {
    float w = (float)k * PIoR;
    float sn, cs;
    __sincosf(w * r, &sn, &cs);
    float bess = K0 * sn * inv;
    float dbess = K0 * (w * cs * inv - sn * inv * inv);
    gr += geb[(size_t)e * NB + k - 1] * (dbess * env + bess * denv);
  }
  float gdu = gux * x + guy * y + guz * z;
  float gvx = gr * x + (gux - gdu * x) * inv;
  float gvy = gr * y + (guy - gdu * y) * inv;
  float gvz = gr * z + (guz - gdu * z) * inv;
  int snd = ei[e], rcv = ei[E + e];
  // force = -dL/dpos ; dL/dpos[rcv] += g_vec ; dL/dpos[snd] -= g_vec
  atomicAdd(&forces[snd * 3 + 0], gvx);
  atomicAdd(&forces[snd * 3 + 1], gvy);
  atomicAdd(&forces[snd * 3 + 2], gvz);
  atomicAdd(&forces[rcv * 3 + 0], -gvx);
  atomicAdd(&forces[rcv * 3 + 1], -gvy);
  atomicAdd(&forces[rcv * 3 + 2], -gvz);
}

// ---------------------------------------------------------------- launcher
extern "C" void kernel_launch(void* const* d_in, const int* in_sizes, int n_in,
                              void* d_out, int out_size, void* d_ws, size_t ws_size,
                              hipStream_t stream) {
  (void)n_in; (void)ws_size;
  const float* pos    = (const float*)d_in[0];
  const float* attrs  = (const float*)d_in[1];
  const float* shifts = (const float*)d_in[2];
  const int*   ei     = (const int*)d_in[3];
  const int*   batch  = (const int*)d_in[4];
  const float* W_emb  = (const float*)d_in[6];
  const float* ae     = (const float*)d_in[7];
  const float* W_r    = (const float*)d_in[8];
  const float* w_sh   = (const float*)d_in[9];
  const float* W_out  = (const float*)d_in[10];
  const float* W_skip = (const float*)d_in[11];
  const float* w_read = (const float*)d_in[12];

  const int N = in_sizes[0] / 3;
  const int E = in_sizes[3] / 2;
  const int G = out_size - 3 * N;

  float* energy = (float*)d_out;
  float* forces = energy + G;

  float* ws = (float*)d_ws;
  const size_t NF_ = (size_t)N * F;
  float* h0   = ws;
  float* h1   = h0 + NF_;
  float* h2   = h1 + NF_;
  float* aggA = h2 + NF_;        // fwd agg layer0, reused as g_agg0 in bwd
  float* aggB = aggA + NF_;      // fwd agg layer1, reused as g_h1 in bwd
  float* evec = aggB + NF_;      // E*3
  float* er_  = evec + (size_t)3 * E;
  float* eeb  = er_ + (size_t)E; // E*8
  float* es   = eeb + (size_t)NB * E; // E*2
  float* geb  = es + (size_t)2 * E;   // E*8
  float* gsb  = geb + (size_t)NB * E; // E*2
  float* v1   = gsb + (size_t)2 * E;  // F

  const int nodeBlocks = (N + 7) / 8;
  const int edgeWaveBlocks = (E + 7) / 8;
  const int edgeThrBlocks = (E + 255) / 256;
  const int gemmBlocks = (N + 15) / 16;

  // init
  k_zero<<<1024, 256, 0, stream>>>(energy, (size_t)out_size);
  k_zero<<<2048, 256, 0, stream>>>(aggA, NF_);
  k_zero<<<2048, 256, 0, stream>>>(aggB, NF_);

  // forward
  k_node_lin<<<nodeBlocks, 256, 0, stream>>>(attrs, W_emb, h0, ae, batch, energy, N);
  k_edge_geom<<<edgeThrBlocks, 256, 0, stream>>>(pos, shifts, ei, w_sh, evec, er_, eeb,
                                                 es, E);
  k_node_lin<<<nodeBlocks, 256, 0, stream>>>(attrs, W_skip, h1, nullptr, nullptr,
                                             nullptr, N);
  k_msg<<<edgeWaveBlocks, 256, 0, stream>>>(h0, eeb, es, ei, W_r, aggA, E, 0);
  k_gemm_wmma<false, true><<<gemmBlocks, 256, 0, stream>>>(aggA, W_out, h1, h1, N);
  k_readout<<<nodeBlocks, 256, 0, stream>>>(h1, w_read, batch, energy, N);
  k_node_lin<<<nodeBlocks, 256, 0, stream>>>(attrs, W_skip + 10 * F, h2, nullptr,
                                             nullptr, nullptr, N);
  k_msg<<<edgeWaveBlocks, 256, 0, stream>>>(h1, eeb, es, ei, W_r + NB * F, aggB, E, 1);
  k_gemm_wmma<false, true><<<gemmBlocks, 256, 0, stream>>>(aggB, W_out + F * F, h2, h2,
                                                           N);
  k_readout<<<nodeBlocks, 256, 0, stream>>>(h2, w_read + F, batch, energy, N);

  // backward (forces)
  k_v1<<<1, F, 0, stream>>>(W_out + F * F, w_read + F, v1);
  k_ginit<<<2048, 256, 0, stream>>>(aggB, w_read, NF_);  // g_h1 = w_read[0]
  k_edge_bwd<1><<<edgeWaveBlocks, 256, 0, stream>>>(h1, eeb, es, ei, W_r + NB * F, v1,
                                                    nullptr, aggB, geb, gsb, E);
  k_gemm_wmma<true, false><<<gemmBlocks, 256, 0, stream>>>(aggB, W_out, nullptr, aggA,
                                                           N);  // g_agg0 = g_h1 @ W0^T
  k_edge_bwd<0><<<edgeWaveBlocks, 256, 0, stream>>>(h0, eeb, es, ei, W_r, nullptr, aggA,
                                                    nullptr, geb, gsb, E);
  k_geom_bwd<<<edgeThrBlocks, 256, 0, stream>>>(evec, er_, geb, gsb, w_sh, ei, forces,
                                                E);
}